// HET_HGTLayerHetero_71365176590768
// MI455X (gfx1250) — compile-verified
//
#include <hip/hip_runtime.h>
#include <math.h>
#include <float.h>

typedef __attribute__((ext_vector_type(2))) float v2f;
typedef __attribute__((ext_vector_type(8))) float v8f;

// Global-address-space float for B-matrix fragment loads -> global_load_b32.
typedef __attribute__((address_space(1))) const float GF;
// LDS address space for async-load destinations.
typedef __attribute__((address_space(3))) const char LDSC;

// ---------------------------------------------------------------------------
// WMMA helper: D = A(16x4 f32) x B(4x16 f32) + C, full f32 precision.
// ---------------------------------------------------------------------------
__device__ inline v8f wmma4(v2f a, v2f b, v8f c) {
  return __builtin_amdgcn_wmma_f32_16x16x4_f32(
      /*neg_a=*/false, a, /*neg_b=*/false, b,
      /*c_mod=*/(short)0, c, /*reuse_a=*/false, /*reuse_b=*/false);
}

// One 16x16 output tile of A(16x64) @ W(64x64)[:, n0:n0+16] using 16 WMMA steps.
// ar[] holds this lane's preloaded A-fragment elements.
__device__ inline v8f gemm_ntile(const float* ar, GF* W, int n0, int colB, int kb2) {
  v8f c = {};
  #pragma unroll
  for (int kk = 0; kk < 16; ++kk) {
    v2f a, b;
    a.x = ar[2 * kk];
    a.y = ar[2 * kk + 1];
    b.x = W[(4 * kk + kb2) * 64 + n0 + colB];
    b.y = W[(4 * kk + kb2 + 1) * 64 + n0 + colB];
    c = wmma4(a, b, c);
  }
  return c;
}

// ---------------------------------------------------------------------------
// Async global->LDS row gather (GLOBAL_LOAD_ASYNC_TO_LDS_B128, ASYNCcnt).
// ---------------------------------------------------------------------------
__device__ inline unsigned lds_off(const void* p) {
  return (unsigned)(unsigned long long)(LDSC*)p;
}
__device__ inline void async_ld_b128(unsigned ldsoff, const float* g) {
  asm volatile("global_load_async_to_lds_b128 %0, %1, off"
               :: "v"(ldsoff), "v"(g)
               : "memory");
}
__device__ inline void wait_async0() {
  asm volatile("s_wait_asynccnt 0" ::: "memory");
}
// Each lane copies 32 consecutive floats (half a row) via 8 async B128 ops.
__device__ inline void gather_half_row(float* ldsdst, const float* gsrc) {
  #pragma unroll
  for (int i = 0; i < 8; ++i)
    async_ld_b128(lds_off(ldsdst + 4 * i), gsrc + 4 * i);
}
__device__ inline void zero_half_row(float* ldsdst) {
  float4 z = {0, 0, 0, 0};
  #pragma unroll
  for (int i = 0; i < 8; ++i) ((float4*)ldsdst)[i] = z;
}

// Order-monotone float <-> uint encoding for atomicMax-based segment max.
__device__ inline unsigned enc_f32(float f) {
  int i = __float_as_int(f);
  unsigned u = (unsigned)i;
  return (i < 0) ? ~u : (u | 0x80000000u);
}
__device__ inline float dec_f32(unsigned k) {
  unsigned u = (k & 0x80000000u) ? (k & 0x7FFFFFFFu) : ~k;
  return __int_as_float((int)u);
}

// ---------------------------------------------------------------------------
// Init: zero agg & den, mkey = enc(-FLT_MAX), zero relation counters.
// ---------------------------------------------------------------------------
__global__ void init_kernel(float* agg, float* den, unsigned* mkey, int* cnt, int N) {
  int i = blockIdx.x * blockDim.x + threadIdx.x;
  if (i < N * 64) agg[i] = 0.0f;
  if (i < N) { den[i] = 0.0f; mkey[i] = enc_f32(-FLT_MAX); }
  if (i < 8) cnt[i] = 0;
}

// ---------------------------------------------------------------------------
// Edge bucketing by relation type (8 relations).
// ---------------------------------------------------------------------------
__global__ void count_kernel(const int* __restrict__ et, int* cnt, int E) {
  int e = blockIdx.x * blockDim.x + threadIdx.x;
  if (e < E) atomicAdd(&cnt[et[e]], 1);
}

__global__ void scan_kernel(const int* __restrict__ cnt, int* off, int* cursor, int* tileoff) {
  if (blockIdx.x == 0 && threadIdx.x == 0) {
    int o = 0, to = 0;
    for (int r = 0; r < 8; ++r) {
      off[r] = o; cursor[r] = o; tileoff[r] = to;
      o += cnt[r];
      to += (cnt[r] + 15) >> 4;
    }
    off[8] = o; tileoff[8] = to;
  }
}

__global__ void fill_kernel(const int* __restrict__ et, int* cursor, int* elist, int E) {
  int e = blockIdx.x * blockDim.x + threadIdx.x;
  if (e < E) {
    int pos = atomicAdd(&cursor[et[e]], 1);
    elist[pos] = e;
  }
}

// ---------------------------------------------------------------------------
// Per-node-type projections: k/q/v = h @ {k,q,v}_lin[node_type].
// One wave per 16-node tile; masked accumulation over the 4 types (WMMA).
// ---------------------------------------------------------------------------
__global__ __launch_bounds__(32)
void proj_kernel(const float* __restrict__ h, const int* __restrict__ ntype,
                 const float* __restrict__ klin, const float* __restrict__ qlin,
                 const float* __restrict__ vlin,
                 float* __restrict__ kb, float* __restrict__ qb, float* __restrict__ vb,
                 int N) {
  __shared__ __align__(16) float A[16][68];
  __shared__ int ty[16];
  int l = threadIdx.x;
  int m0 = blockIdx.x * 16;

  if (l < 16) {
    int g = m0 + l;
    ty[l] = (g < N) ? ntype[g] : -1;
  }
  {
    int row = l >> 1;
    int half = (l & 1) * 32;
    int g = m0 + row;
    if (g < N) gather_half_row(&A[row][half], h + (size_t)g * 64 + half);
    else       zero_half_row(&A[row][half]);
  }
  wait_async0();
  __syncthreads();

  int rowA  = l & 15;
  int kb2   = ((l >> 4) & 1) * 2;   // K sub-offset: 0 for lanes 0-15, 2 for 16-31
  int colB  = l & 15;
  int rbase = (l >> 4) * 8;         // C/D rows 0-7 or 8-15

  float ar[32];
  #pragma unroll
  for (int kk = 0; kk < 16; ++kk) {
    ar[2 * kk]     = A[rowA][4 * kk + kb2];
    ar[2 * kk + 1] = A[rowA][4 * kk + kb2 + 1];
  }

  unsigned tmask = 0;
  for (int r = 0; r < 16; ++r) if (ty[r] >= 0) tmask |= 1u << ty[r];

  GF* Ws[3] = {(GF*)klin, (GF*)qlin, (GF*)vlin};
  float* Os[3] = {kb, qb, vb};

  for (int mtx = 0; mtx < 3; ++mtx) {
    GF* Wb = Ws[mtx];
    float* Ob = Os[mtx];
    for (int n0 = 0; n0 < 64; n0 += 16) {
      float res[8];
      #pragma unroll
      for (int i = 0; i < 8; ++i) res[i] = 0.0f;
      for (int t = 0; t < 4; ++t) {
        if (!(tmask & (1u << t))) continue;
        v8f c = gemm_ntile(ar, Wb + (size_t)t * 64 * 64, n0, colB, kb2);
        #pragma unroll
        for (int i = 0; i < 8; ++i)
          if (ty[rbase + i] == t) res[i] = c[i];
      }
      #pragma unroll
      for (int i = 0; i < 8; ++i) {
        int g = m0 + rbase + i;
        if (g < N) Ob[(size_t)g * 64 + n0 + colB] = res[i];
      }
    }
  }
}

// ---------------------------------------------------------------------------
// Attention scores per relation bucket: 16 edges/tile.
// T = q[dst] @ rel_att[r] (WMMA), score = <T_row, k[src]>, atomicMax seg-max.
// ---------------------------------------------------------------------------
__global__ __launch_bounds__(32)
void score_kernel(const float* __restrict__ qb, const float* __restrict__ kb,
                  const float* __restrict__ rel_att, const float* __restrict__ rel_pri,
                  const int* __restrict__ src, const int* __restrict__ dst,
                  const int* __restrict__ elist, const int* __restrict__ cnt,
                  const int* __restrict__ off, const int* __restrict__ tileoff,
                  float* __restrict__ logits, unsigned* __restrict__ mkey) {
  __shared__ __align__(16) float Q[16][68];
  __shared__ __align__(16) float T[16][68];
  __shared__ __align__(16) float Kr[16][68];
  __shared__ int eid[16], didx[16], sidx[16];
  int l = threadIdx.x;
  int b = blockIdx.x;

  int r = -1, lt = 0;
  for (int i = 0; i < 8; ++i) {
    if (b >= tileoff[i] && b < tileoff[i + 1]) { r = i; lt = b - tileoff[i]; break; }
  }
  if (r < 0) return;
  int nE = cnt[r];

  if (l < 16) {
    int j = lt * 16 + l;
    if (j < nE) {
      int e = elist[off[r] + j];
      eid[l] = e; didx[l] = dst[e]; sidx[l] = src[e];
    } else { eid[l] = -1; didx[l] = -1; sidx[l] = -1; }
  }
  __syncthreads();
  {
    int row = l >> 1;
    int half = (l & 1) * 32;
    int d = didx[row], s = sidx[row];
    if (d >= 0) {
      gather_half_row(&Q[row][half],  qb + (size_t)d * 64 + half);
      gather_half_row(&Kr[row][half], kb + (size_t)s * 64 + half);
    } else {
      zero_half_row(&Q[row][half]);
      zero_half_row(&Kr[row][half]);
    }
  }
  wait_async0();
  __syncthreads();

  int rowA  = l & 15;
  int kb2   = ((l >> 4) & 1) * 2;
  int colB  = l & 15;
  int rbase = (l >> 4) * 8;
  GF* W = (GF*)rel_att + (size_t)r * 64 * 64;

  float ar[32];
  #pragma unroll
  for (int kk = 0; kk < 16; ++kk) {
    ar[2 * kk]     = Q[rowA][4 * kk + kb2];
    ar[2 * kk + 1] = Q[rowA][4 * kk + kb2 + 1];
  }
  for (int n0 = 0; n0 < 64; n0 += 16) {
    v8f c = gemm_ntile(ar, W, n0, colB, kb2);
    #pragma unroll
    for (int i = 0; i < 8; ++i) T[rbase + i][n0 + colB] = c[i];
  }
  __syncthreads();

  if (l < 16 && eid[l] >= 0) {
    float s = 0.0f;
    #pragma unroll
    for (int cidx = 0; cidx < 64; ++cidx) s += T[l][cidx] * Kr[l][cidx];
    float logit = s * rel_pri[r] * 0.125f;  // rel_pri / sqrt(64)
    logits[eid[l]] = logit;
    atomicMax(&mkey[didx[l]], enc_f32(logit));
  }
}

// ---------------------------------------------------------------------------
// exp(logit - max) + denominator segment-sum.
// ---------------------------------------------------------------------------
__global__ void expden_kernel(const float* __restrict__ logits, const int* __restrict__ dst,
                              const unsigned* __restrict__ mkey,
                              float* __restrict__ exb, float* __restrict__ den, int E) {
  int e = blockIdx.x * blockDim.x + threadIdx.x;
  if (e < E) {
    int d = dst[e];
    float ex = expf(logits[e] - dec_f32(mkey[d]));
    exb[e] = ex;
    atomicAdd(&den[d], ex);
  }
}

// ---------------------------------------------------------------------------
// Messages per relation bucket + softmax-weighted scatter-add to agg[dst].
// ---------------------------------------------------------------------------
__global__ __launch_bounds__(32)
void msg_kernel(const float* __restrict__ vb, const float* __restrict__ rel_msg,
                const int* __restrict__ src, const int* __restrict__ dst,
                const int* __restrict__ elist, const int* __restrict__ cnt,
                const int* __restrict__ off, const int* __restrict__ tileoff,
                const float* __restrict__ exb, const float* __restrict__ den,
                float* __restrict__ agg) {
  __shared__ __align__(16) float V[16][68];
  __shared__ __align__(16) float T[16][68];
  __shared__ int eid[16], didx[16], sidx[16];
  __shared__ float alpha[16];
  int l = threadIdx.x;
  int b = blockIdx.x;

  int r = -1, lt = 0;
  for (int i = 0; i < 8; ++i) {
    if (b >= tileoff[i] && b < tileoff[i + 1]) { r = i; lt = b - tileoff[i]; break; }
  }
  if (r < 0) return;
  int nE = cnt[r];

  if (l < 16) {
    int j = lt * 16 + l;
    if (j < nE) {
      int e = elist[off[r] + j];
      eid[l] = e; didx[l] = dst[e]; sidx[l] = src[e];
    } else { eid[l] = -1; didx[l] = -1; sidx[l] = -1; }
  }
  __syncthreads();
  {
    int row = l >> 1;
    int half = (l & 1) * 32;
    int s = sidx[row];
    if (s >= 0) gather_half_row(&V[row][half], vb + (size_t)s * 64 + half);
    else        zero_half_row(&V[row][half]);
  }
  wait_async0();
  __syncthreads();

  int rowA  = l & 15;
  int kb2   = ((l >> 4) & 1) * 2;
  int colB  = l & 15;
  int rbase = (l >> 4) * 8;
  GF* W = (GF*)rel_msg + (size_t)r * 64 * 64;

  float ar[32];
  #pragma unroll
  for (int kk = 0; kk < 16; ++kk) {
    ar[2 * kk]     = V[rowA][4 * kk + kb2];
    ar[2 * kk + 1] = V[rowA][4 * kk + kb2 + 1];
  }
  for (int n0 = 0; n0 < 64; n0 += 16) {
    v8f c = gemm_ntile(ar, W, n0, colB, kb2);
    #pragma unroll
    for (int i = 0; i < 8; ++i) T[rbase + i][n0 + colB] = c[i];
  }

  if (l < 16) {
    float a = 0.0f;
    if (eid[l] >= 0) {
      float dv = den[didx[l]];
      a = exb[eid[l]] / (dv == 0.0f ? 1.0f : dv);
    }
    alpha[l] = a;
  }
  __syncthreads();

  for (int j = 0; j < 16; ++j) {
    if (eid[j] < 0) continue;
    float av = alpha[j];
    size_t dbase = (size_t)didx[j] * 64;
    atomicAdd(&agg[dbase + l],      av * T[j][l]);
    atomicAdd(&agg[dbase + l + 32], av * T[j][l + 32]);
  }
}

// ---------------------------------------------------------------------------
// Output: out = agg @ (sigmoid(skip[t]) * a_lin[t]), per node type (WMMA).
// ---------------------------------------------------------------------------
__global__ __launch_bounds__(32)
void out_kernel(const float* __restrict__ agg, const int* __restrict__ ntype,
                const float* __restrict__ alin, const float* __restrict__ skip,
                float* __restrict__ out, int N) {
  __shared__ __align__(16) float A[16][68];
  __shared__ int ty[16];
  int l = threadIdx.x;
  int m0 = blockIdx.x * 16;

  if (l < 16) {
    int g = m0 + l;
    ty[l] = (g < N) ? ntype[g] : -1;
  }
  {
    int row = l >> 1;
    int half = (l & 1) * 32;
    int g = m0 + row;
    if (g < N) gather_half_row(&A[row][half], agg + (size_t)g * 64 + half);
    else       zero_half_row(&A[row][half]);
  }
  wait_async0();
  __syncthreads();

  int rowA  = l & 15;
  int kb2   = ((l >> 4) & 1) * 2;
  int colB  = l & 15;
  int rbase = (l >> 4) * 8;

  float ar[32];
  #pragma unroll
  for (int kk = 0; kk < 16; ++kk) {
    ar[2 * kk]     = A[rowA][4 * kk + kb2];
    ar[2 * kk + 1] = A[rowA][4 * kk + kb2 + 1];
  }

  unsigned tmask = 0;
  for (int r = 0; r < 16; ++r) if (ty[r] >= 0) tmask |= 1u << ty[r];

  for (int n0 = 0; n0 < 64; n0 += 16) {
    float res[8];
    #pragma unroll
    for (int i = 0; i < 8; ++i) res[i] = 0.0f;
    for (int t = 0; t < 4; ++t) {
      if (!(tmask & (1u << t))) continue;
      float sg = 1.0f / (1.0f + expf(-skip[t]));
      GF* W = (GF*)alin + (size_t)t * 64 * 64;
      v8f c = {};
      #pragma unroll
      for (int kk = 0; kk < 16; ++kk) {
        v2f a, b;
        a.x = ar[2 * kk];
        a.y = ar[2 * kk + 1];
        b.x = sg * W[(4 * kk + kb2) * 64 + n0 + colB];
        b.y = sg * W[(4 * kk + kb2 + 1) * 64 + n0 + colB];
        c = wmma4(a, b, c);
      }
      #pragma unroll
      for (int i = 0; i < 8; ++i)
        if (ty[rbase + i] == t) res[i] = c[i];
    }
    #pragma unroll
    for (int i = 0; i < 8; ++i) {
      int g = m0 + rbase + i;
      if (g < N) out[(size_t)g * 64 + n0 + colB] = res[i];
    }
  }
}

// ---------------------------------------------------------------------------
// Host launcher
// ---------------------------------------------------------------------------
extern "C" void kernel_launch(void* const* d_in, const int* in_sizes, int n_in,
                              void* d_out, int out_size, void* d_ws, size_t ws_size,
                              hipStream_t stream) {
  const float* h     = (const float*)d_in[0];
  const int*   ntype = (const int*)d_in[1];
  const int*   src   = (const int*)d_in[2];
  const int*   dst   = (const int*)d_in[3];
  const int*   et    = (const int*)d_in[4];
  const float* klin  = (const float*)d_in[5];
  const float* qlin  = (const float*)d_in[6];
  const float* vlin  = (const float*)d_in[7];
  const float* alin  = (const float*)d_in[8];
  const float* ratt  = (const float*)d_in[9];
  const float* rmsg  = (const float*)d_in[10];
  const float* rpri  = (const float*)d_in[11];
  const float* skip  = (const float*)d_in[12];
  float* out = (float*)d_out;

  const int N = in_sizes[1];
  const int E = in_sizes[2];

  char* base = (char*)d_ws;
  size_t cur = 0;
  auto wsalloc = [&](size_t bytes) -> void* {
    void* p = base + cur;
    cur += (bytes + 255) & ~(size_t)255;
    return p;
  };
  float*    qb      = (float*)wsalloc((size_t)N * 64 * 4);
  float*    kb      = (float*)wsalloc((size_t)N * 64 * 4);
  float*    vb      = (float*)wsalloc((size_t)N * 64 * 4);
  float*    agg     = (float*)wsalloc((size_t)N * 64 * 4);
  float*    logits  = (float*)wsalloc((size_t)E * 4);
  float*    exb     = (float*)wsalloc((size_t)E * 4);
  unsigned* mkey    = (unsigned*)wsalloc((size_t)N * 4);
  float*    den     = (float*)wsalloc((size_t)N * 4);
  int*      elist   = (int*)wsalloc((size_t)E * 4);
  int*      cnt     = (int*)wsalloc(8 * 4);
  int*      off     = (int*)wsalloc(9 * 4);
  int*      cursor  = (int*)wsalloc(8 * 4);
  int*      tileoff = (int*)wsalloc(9 * 4);

  const int nodeTiles = (N + 15) / 16;
  const int edgeTilesMax = (E + 15) / 16 + 8;

  init_kernel<<<(N * 64 + 255) / 256, 256, 0, stream>>>(agg, den, mkey, cnt, N);
  count_kernel<<<(E + 255) / 256, 256, 0, stream>>>(et, cnt, E);
  scan_kernel<<<1, 32, 0, stream>>>(cnt, off, cursor, tileoff);
  fill_kernel<<<(E + 255) / 256, 256, 0, stream>>>(et, cursor, elist, E);

  proj_kernel<<<nodeTiles, 32, 0, stream>>>(h, ntype, klin, qlin, vlin, kb, qb, vb, N);

  score_kernel<<<edgeTilesMax, 32, 0, stream>>>(qb, kb, ratt, rpri, src, dst,
                                                elist, cnt, off, tileoff, logits, mkey);
  expden_kernel<<<(E + 255) / 256, 256, 0, stream>>>(logits, dst, mkey, exb, den, E);
  msg_kernel<<<edgeTilesMax, 32, 0, stream>>>(vb, rmsg, src, dst, elist, cnt, off,
                                              tileoff, exb, den, agg);

  out_kernel<<<nodeTiles, 32, 0, stream>>>(agg, ntype, alin, skip, out, N);
}